// ConSegLoss_44822278701825
// MI455X (gfx1250) — compile-verified
//
#include <hip/hip_runtime.h>

// Problem constants from the reference (fixed shapes).
#define BATCH 16
#define KPTS  17
#define HH    256
#define WW    256
#define HW    65536      // HH*WW
#define NPRED 4
#define WSUM  2.952f     // 0.8^3 + 0.8^2 + 0.8 + 1
#define EPS   1e-6f
#define THRES 0.95f

typedef __attribute__((ext_vector_type(2))) float v2f;
typedef __attribute__((ext_vector_type(8))) float v8f;

// Wave32 sum-reduction via V_WMMA_F32_16X16X4_F32 with B = ones.
// A layout (ISA 7.12.2, 32-bit A 16x4): lanes 0-15 hold K=0/K=1 in a.x/a.y,
// lanes 16-31 hold K=2/K=3. With per-lane partial in a.x and 0 in a.y:
//   D[m][n] = p[m] + p[m+16]  (independent of n, so B's layout is irrelevant).
// C/D layout: lane n<16 holds rows 0..7 in c[0..7]; lane n+16 holds rows 8..15.
// So sum(c[0..7]) gives half the total per lane; shfl_xor(16) + add completes it.
__device__ __forceinline__ float wave_sum_wmma(float partial) {
    v2f a; a.x = partial; a.y = 0.0f;
    v2f bones; bones.x = 1.0f; bones.y = 1.0f;
    v8f c = {};
    c = __builtin_amdgcn_wmma_f32_16x16x4_f32(
            /*neg_a=*/false, a, /*neg_b=*/false, bones,
            /*c_mod=*/(short)0, c, /*reuse_a=*/false, /*reuse_b=*/false);
    float s = c[0] + c[1] + c[2] + c[3] + c[4] + c[5] + c[6] + c[7];
    s += __shfl_xor(s, 16, 32);
    return s;   // full wave sum, valid in every lane
}

__global__ __launch_bounds__(256)
void conseg_main(const float* __restrict__ masks,
                 const float* __restrict__ pred_flows,
                 const int*   __restrict__ skls,
                 const float* __restrict__ flows,
                 float* __restrict__ out,          // [0]=loss (written later), [1..]=mask_gt
                 float* __restrict__ block_sums) {
    __shared__ int   s_sx[KPTS], s_sy[KPTS];
    __shared__ float s_f0[KPTS], s_f1[KPTS], s_kn[KPTS];
    __shared__ float s_wave[8];

    const int b   = blockIdx.y;
    const int tid = threadIdx.x;

    // Per-batch keypoint gather: kp_flow[b,k,:] = flows[b,:,sy,sx], plus its norm.
    if (tid < KPTS) {
        int sx = skls[(b * KPTS + tid) * 2 + 0];
        int sy = skls[(b * KPTS + tid) * 2 + 1];
        s_sx[tid] = sx; s_sy[tid] = sy;
        float f0 = flows[((b * 2 + 0) << 16) + (sy << 8) + sx];
        float f1 = flows[((b * 2 + 1) << 16) + (sy << 8) + sx];
        s_f0[tid] = f0; s_f1[tid] = f1;
        s_kn[tid] = sqrtf(f0 * f0 + f1 * f1);
    }
    __syncthreads();

    const int p = blockIdx.x * 256 + tid;   // pixel index within batch plane
    const int y = p >> 8;
    const int x = p & 255;

    // which_kp: argmin of integer squared distance (exact; first-min wins like jnp.argmin).
    int best = 0, bestd = 0x7fffffff;
    #pragma unroll
    for (int k = 0; k < KPTS; ++k) {
        int dx = x - s_sx[k];
        int dy = y - s_sy[k];
        int d  = dx * dx + dy * dy;
        if (d < bestd) { bestd = d; best = k; }
    }
    const float kf0 = s_f0[best], kf1 = s_f1[best], kn = s_kn[best];

    const float xm   = masks[(b << 16) + p];
    // BCE base term shared by all i: max(x,0) + softplus(-|x|)
    const float base = fmaxf(xm, 0.0f) + log1pf(expf(-fabsf(xm)));

    const float wts[NPRED] = {0.512f, 0.64f, 0.8f, 1.0f};  // gamma^(N-1-i)
    float zsum = 0.0f, z_last = 0.0f;
    #pragma unroll
    for (int i = 0; i < NPRED; ++i) {
        const float* pf = pred_flows + (((size_t)((i * BATCH + b) * 2)) << 16) + p;
        float p0 = pf[0];
        float p1 = pf[HW];
        float dot = p0 * kf0 + p1 * kf1;
        float pn  = sqrtf(p0 * p0 + p1 * p1);
        float sim = dot / (pn * kn + EPS);
        float z   = (sim > THRES) ? 1.0f : 0.0f;
        zsum += wts[i] * z;
        if (i == NPRED - 1) z_last = z;
    }
    out[1 + (b << 16) + p] = z_last;   // mask_gt of last prediction

    // Weighted per-pixel loss contribution (pre-division by B*H*W).
    float partial = WSUM * base - xm * zsum;

    // Block reduction: WMMA wave sum -> LDS -> thread 0.
    float s = wave_sum_wmma(partial);
    const int lane = tid & 31, wave = tid >> 5;
    if (lane == 0) s_wave[wave] = s;
    __syncthreads();
    if (tid == 0) {
        float t = 0.0f;
        #pragma unroll
        for (int w = 0; w < 8; ++w) t += s_wave[w];
        block_sums[blockIdx.y * gridDim.x + blockIdx.x] = t;
    }
}

// Deterministic final reduction of 4096 block sums -> out[0].
__global__ __launch_bounds__(256)
void conseg_reduce(const float* __restrict__ block_sums, float* __restrict__ out) {
    __shared__ float s_wave[8];
    const int tid = threadIdx.x;
    float acc = 0.0f;
    #pragma unroll
    for (int j = 0; j < 16; ++j) acc += block_sums[tid + j * 256];
    float s = wave_sum_wmma(acc);
    const int lane = tid & 31, wave = tid >> 5;
    if (lane == 0) s_wave[wave] = s;
    __syncthreads();
    if (tid == 0) {
        float t = 0.0f;
        #pragma unroll
        for (int w = 0; w < 8; ++w) t += s_wave[w];
        out[0] = t * (1.0f / (float)(BATCH * HW));   // the mean over B*1*H*W
    }
}

extern "C" void kernel_launch(void* const* d_in, const int* in_sizes, int n_in,
                              void* d_out, int out_size, void* d_ws, size_t ws_size,
                              hipStream_t stream) {
    const float* masks      = (const float*)d_in[0];   // (16,1,256,256) f32
    const float* pred_flows = (const float*)d_in[1];   // (4,16,2,256,256) f32
    const int*   skls       = (const int*)  d_in[2];   // (16,17,2) i32
    const float* flows      = (const float*)d_in[3];   // (16,2,256,256) f32
    float* out        = (float*)d_out;                 // [0]=loss, [1..]=mask_gt
    float* block_sums = (float*)d_ws;                  // 4096 floats of scratch

    dim3 grid(HW / 256, BATCH);   // 256 x 16 = 4096 blocks, each = 256 pixels of one batch
    conseg_main<<<grid, 256, 0, stream>>>(masks, pred_flows, skls, flows, out, block_sums);
    conseg_reduce<<<1, 256, 0, stream>>>(block_sums, out);
}